// testWarp_13666585936557
// MI455X (gfx1250) — compile-verified
//
#include <hip/hip_runtime.h>
#include <hip/hip_bf16.h>

// Bilinear horizontal warp (disparity warp), B=8, C=16, H=512, W=1024, fp32.
//
// Strategy (MI455X / gfx1250):
//   - one workgroup per (b,h) row; async-DMA all 16 channel rows (64 KB) from
//     global into LDS with GLOBAL_LOAD_ASYNC_TO_LDS_B128 (ASYNCcnt path),
//   - s_wait_asynccnt 0 + workgroup barrier,
//   - each thread computes x0/x1/weights once per w (disp shared across C),
//     then gathers from LDS for all 16 channels and streams coalesced stores.
//   - output stores and disp loads are non-temporal (touch-once streaming;
//     don't churn the 192 MB L2 with 272 MB of write/read-once data).
// Memory-bound: ~528 MB total traffic -> ~23 us floor at 23.3 TB/s.

#define C_ 16
#define H_ 512
#define W_ 1024
#define BDIM 256

// Exact pointee type required by the async-LDS builtin (probe-confirmed):
// int __attribute__((vector_size(16))).
typedef int v4i __attribute__((vector_size(16)));

// ---- async global -> LDS copy of 16 bytes per lane --------------------------
#if defined(__HIP_DEVICE_COMPILE__) && __has_builtin(__builtin_amdgcn_global_load_async_to_lds_b128)
#define ASYNC_LDS_B128(gp, lp)                                                    \
  __builtin_amdgcn_global_load_async_to_lds_b128(                                 \
      (__attribute__((address_space(1))) v4i*)(unsigned long long)(gp),           \
      (__attribute__((address_space(3))) v4i*)(unsigned)(unsigned long long)(lp), \
      0, 0)
#else
// GV mode: 64-bit global address in a VGPR pair, LDS byte address in a VGPR.
#define ASYNC_LDS_B128(gp, lp)                                                    \
  asm volatile("global_load_async_to_lds_b128 %0, %1, off"                        \
               :: "v"((unsigned)(unsigned long long)(lp)),                        \
                  "v"((unsigned long long)(gp))                                   \
               : "memory")
#endif

#if defined(__HIP_DEVICE_COMPILE__) && __has_builtin(__builtin_amdgcn_s_wait_asynccnt)
#define WAIT_ASYNC0() __builtin_amdgcn_s_wait_asynccnt(0)
#else
#define WAIT_ASYNC0() asm volatile("s_wait_asynccnt 0x0" ::: "memory")
#endif

__global__ __launch_bounds__(BDIM) void warp_bilinear_kernel(
    const float* __restrict__ input,   // [B,C,H,W]
    const float* __restrict__ disp,    // [B,1,H,W]
    float* __restrict__ out)           // [B,C,H,W]
{
  __shared__ float smem[C_ * W_];      // 64 KB: 16 channel rows of 1024 floats

  const int tid = threadIdx.x;         // 0..255 (8 wave32's)
  const int bh  = blockIdx.x;          // 0..B*H-1
  const int b   = bh >> 9;             // H_ == 512
  const int h   = bh & (H_ - 1);

  // ---- Phase 1: stage the 16 channel rows into LDS (streaming b128) --------
  // Each of the 256 threads copies 16 bytes per channel: 256*16B = 4 KB = row.
  {
    const float* g0 = input + (((size_t)b * C_) * H_ + (size_t)h) * W_;
#pragma unroll
    for (int c = 0; c < C_; ++c) {
      const float* gp = g0 + (size_t)c * (H_ * (size_t)W_) + (size_t)tid * 4;
      const float* lp = &smem[c * W_ + tid * 4];
      ASYNC_LDS_B128(gp, lp);
    }
  }
  WAIT_ASYNC0();       // this wave's 16 async copies landed in LDS
  __syncthreads();     // all 8 waves' copies landed

  // ---- Phase 2: bilinear gather from LDS, NT coalesced stores --------------
  const float* drow = disp + ((size_t)b * H_ + (size_t)h) * W_;
  float* orow0 = out + (((size_t)b * C_) * H_ + (size_t)h) * W_;  // channel 0
  const size_t cstride = (size_t)H_ * W_;                          // channel stride

#pragma unroll
  for (int j = 0; j < W_ / BDIM; ++j) {       // 4 iterations
    const int w = tid + j * BDIM;             // lane-contiguous w
    float x = (float)w + __builtin_nontemporal_load(&drow[w]);
    x = fminf(fmaxf(x, 0.0f), (float)(W_ - 1));
    const float x0f = floorf(x);
    const float x1f = fminf(x0f + 1.0f, (float)(W_ - 1));
    const int x0 = (int)x0f;
    const int x1 = (int)x1f;
    const float wl = x1f - x;   // both 0 at the right clamp (matches reference)
    const float wr = x - x0f;

#pragma unroll
    for (int c = 0; c < C_; ++c) {
      const float pl = smem[c * W_ + x0];
      const float pr = smem[c * W_ + x1];
      __builtin_nontemporal_store(fmaf(wl, pl, wr * pr),
                                  &orow0[(size_t)c * cstride + w]);
    }
  }
}

extern "C" void kernel_launch(void* const* d_in, const int* in_sizes, int n_in,
                              void* d_out, int out_size, void* d_ws, size_t ws_size,
                              hipStream_t stream) {
  const float* input = (const float*)d_in[0];  // [8,16,512,1024] f32
  const float* disp  = (const float*)d_in[1];  // [8,1,512,1024] f32
  float* out = (float*)d_out;                  // [8,16,512,1024] f32

  const int B = 8;
  dim3 grid(B * H_);      // 4096 workgroups, one per (b,h) row
  dim3 block(BDIM);       // 8 waves of 32
  warp_bilinear_kernel<<<grid, block, 0, stream>>>(input, disp, out);
}